// ConditionEmbeddingLayer_82789789598114
// MI455X (gfx1250) — compile-verified
//
#include <hip/hip_runtime.h>

#define N_GENES 20000
#define DD      256
#define NNZ_E   640000
#define BB      16384
#define MAXG    5

typedef __attribute__((ext_vector_type(2))) float v2f;
typedef __attribute__((ext_vector_type(8))) float v8f;

// ---------------- zero an int buffer ----------------
__global__ void zero_i32(int* __restrict__ p, int n) {
  int i = blockIdx.x * blockDim.x + threadIdx.x;
  if (i < n) p[i] = 0;
}

// ---------------- CSR build: count edges per row ----------------
__global__ void count_edges(const int* __restrict__ rows, int* __restrict__ cnt) {
  int e = blockIdx.x * blockDim.x + threadIdx.x;
  if (e < NNZ_E) atomicAdd(&cnt[rows[e]], 1);
}

// ---------------- CSR build: exclusive scan of 20000 counts (single block) ----------------
__global__ void scan_rows(const int* __restrict__ cnt, int* __restrict__ rowptr,
                          int* __restrict__ cursor) {
  __shared__ int partial[256];
  const int t = threadIdx.x;
  const int chunk = (N_GENES + 255) / 256;   // 79
  const int base = t * chunk;
  int s = 0;
  for (int i = 0; i < chunk; ++i) {
    int idx = base + i;
    if (idx < N_GENES) s += cnt[idx];
  }
  partial[t] = s;
  __syncthreads();
  if (t == 0) {
    int acc = 0;
    for (int i = 0; i < 256; ++i) { int v = partial[i]; partial[i] = acc; acc += v; }
  }
  __syncthreads();
  int acc = partial[t];
  for (int i = 0; i < chunk; ++i) {
    int idx = base + i;
    if (idx < N_GENES) { rowptr[idx] = acc; cursor[idx] = acc; acc += cnt[idx]; }
  }
}

// ---------------- CSR build: scatter edges into row-sorted order ----------------
__global__ void scatter_edges(const int* __restrict__ rows, const int* __restrict__ cols,
                              const float* __restrict__ vals, int* __restrict__ cursor,
                              int* __restrict__ ccol, float* __restrict__ cval) {
  int e = blockIdx.x * blockDim.x + threadIdx.x;
  if (e < NNZ_E) {
    int p = atomicAdd(&cursor[rows[e]], 1);
    ccol[p] = cols[e];
    cval[p] = vals[e];
  }
}

// ---------------- atomic-free row-parallel SpMM: dst[r] = sum_e val_e * src[col_e] ----------------
// One wave per gene row. Edges are loaded 32-at-a-time into registers and broadcast
// across the wave with __shfl (wave32). Each lane accumulates 8 channels in VGPRs;
// the 1KB row gather per edge hits L2 (src ~20.5MB << 192MB L2). Fully overwrites dst.
__global__ void spmm_csr(const int* __restrict__ rowptr, const int* __restrict__ cnt,
                         const int* __restrict__ ccol, const float* __restrict__ cval,
                         const float* __restrict__ src, float* __restrict__ dst) {
  int wid  = (blockIdx.x * blockDim.x + threadIdx.x) >> 5;   // row id
  int lane = threadIdx.x & 31;
  if (wid >= N_GENES) return;
  int start = rowptr[wid];
  int len   = cnt[wid];
  float4 a0 = {0.f, 0.f, 0.f, 0.f};
  float4 a1 = {0.f, 0.f, 0.f, 0.f};
  for (int base = 0; base < len; base += 32) {
    int n = min(32, len - base);
    int   c = 0;
    float v = 0.f;
    if (base + lane < len) {
      c = ccol[start + base + lane];
      v = cval[start + base + lane];
    }
    for (int j = 0; j < n; ++j) {
      int   cj = __shfl(c, j, 32);
      float vj = __shfl(v, j, 32);
      const float4* p = (const float4*)(src + (long long)cj * DD) + lane * 2;
      float4 e0 = p[0];
      float4 e1 = p[1];
      a0.x += vj * e0.x; a0.y += vj * e0.y; a0.z += vj * e0.z; a0.w += vj * e0.w;
      a1.x += vj * e1.x; a1.y += vj * e1.y; a1.z += vj * e1.z; a1.w += vj * e1.w;
    }
  }
  float4* d = (float4*)(dst + (long long)wid * DD) + lane * 2;
  d[0] = a0;
  d[1] = a1;
}

// ---------------- per-sample gather-sum over up to MAXG genes ----------------
__global__ void gather_sum(const int* __restrict__ cond_idx,
                           const int* __restrict__ gidx, const float* __restrict__ gmask,
                           const float* __restrict__ emb, float* __restrict__ summed,
                           float* __restrict__ ngenes) {
  int gid = blockIdx.x * blockDim.x + threadIdx.x;  // BB * 64 threads
  int b  = gid >> 6;
  int c4 = gid & 63;
  if (b >= BB) return;
  int ci = cond_idx[b];
  float4 acc = {0.f, 0.f, 0.f, 0.f};
  float n = 0.f;
#pragma unroll
  for (int g = 0; g < MAXG; ++g) {
    int   gi = gidx[ci * MAXG + g];
    float m  = gmask[ci * MAXG + g];
    n += m;
    if (gi >= 0) {
      float4 e = ((const float4*)(emb + (long long)gi * DD))[c4];
      acc.x += m * e.x; acc.y += m * e.y; acc.z += m * e.z; acc.w += m * e.w;
    }
  }
  ((float4*)(summed + (long long)b * DD))[c4] = acc;
  if (c4 == 0) ngenes[b] = n;
}

// ---------------- f32 WMMA GEMM: C[M x 256] = act(A[M x 256] @ W[256 x 256] + bias) ----------------
// One wave owns a 32x16 output macro-tile (2 stacked 16x16 WMMA tiles sharing B fragments:
// 2 loads/WMMA instead of 3). 8 waves per block. Caller guarantees M % 32 == 0 and
// (M/32*16) % 8 == 0 so every wave has a full tile and EXEC stays all-ones for WMMA.
__global__ void gemm_wmma_f32(const float* __restrict__ A, const float* __restrict__ W,
                              const float* __restrict__ bias, float* __restrict__ Cout,
                              int relu,
                              const float* __restrict__ blend_src,
                              const float* __restrict__ ngenes) {
  const int lane = threadIdx.x & 31;
  const int wave = threadIdx.x >> 5;
  const int tiles_n = DD / 16;                    // 16 column tiles
  int tile = blockIdx.x * 8 + wave;
  int tp = tile / tiles_n;                        // M macro-tile (32 rows)
  int tn = tile - tp * tiles_n;
  int row_base = tp * 32;
  int col_base = tn * 16;
  int m16 = lane & 15;                            // M index (A) / N index (B,C)
  int hi  = lane >> 4;                            // K half selector

  const float* arow0 = A + (long long)(row_base + m16) * DD;
  const float* arow1 = arow0 + 16 * DD;
  const float* bcol  = W + col_base + m16;

  v8f acc0 = {0.f, 0.f, 0.f, 0.f, 0.f, 0.f, 0.f, 0.f};
  v8f acc1 = {0.f, 0.f, 0.f, 0.f, 0.f, 0.f, 0.f, 0.f};
#pragma unroll 4
  for (int kk = 0; kk < DD; kk += 4) {
    int k0 = kk + 2 * hi;
    v2f a0 = *(const v2f*)(arow0 + k0);
    v2f a1 = *(const v2f*)(arow1 + k0);
    v2f b;
    b.x = bcol[(long long)k0 * DD];
    b.y = bcol[(long long)(k0 + 1) * DD];
    acc0 = __builtin_amdgcn_wmma_f32_16x16x4_f32(false, a0, false, b, (short)0, acc0, false, false);
    acc1 = __builtin_amdgcn_wmma_f32_16x16x4_f32(false, a1, false, b, (short)0, acc1, false, false);
  }

  float bv = bias ? bias[col_base + m16] : 0.f;
#pragma unroll
  for (int half = 0; half < 2; ++half) {
    v8f acc = half ? acc1 : acc0;
    int rb = row_base + half * 16;
#pragma unroll
    for (int r = 0; r < 8; ++r) {
      int row = rb + r + 8 * hi;                  // C/D layout: lanes 16-31 -> M = r+8
      float v = acc[r] + bv;
      if (relu) v = fmaxf(v, 0.f);
      if (ngenes) {
        float nn = ngenes[row];
        float s  = blend_src[(long long)row * DD + col_base + m16];
        v = (nn == 0.f) ? 0.f : ((nn == 1.f) ? s : v);
      }
      Cout[(long long)row * DD + col_base + m16] = v;
    }
  }
}

extern "C" void kernel_launch(void* const* d_in, const int* in_sizes, int n_in,
                              void* d_out, int out_size, void* d_ws, size_t ws_size,
                              hipStream_t stream) {
  (void)in_sizes; (void)n_in; (void)out_size; (void)ws_size;
  const int*   cond_idx = (const int*)d_in[0];
  const float* pert     = (const float*)d_in[1];
  const float* gk       = (const float*)d_in[2];   // (2, 256, 256)
  const float* w1       = (const float*)d_in[3];
  const float* b1       = (const float*)d_in[4];
  const float* w2       = (const float*)d_in[5];
  const float* b2       = (const float*)d_in[6];
  const int*   arow     = (const int*)d_in[7];
  const int*   acol     = (const int*)d_in[8];
  const float* avals    = (const float*)d_in[9];
  const int*   cgi      = (const int*)d_in[10];
  const float* cgm      = (const float*)d_in[11];
  float*       out      = (float*)d_out;

  // Workspace layout (4-byte words)
  float* ws     = (float*)d_ws;
  float* aggA   = ws;                                  // N_GENES*DD
  float* embB   = aggA   + (size_t)N_GENES * DD;       // N_GENES*DD
  float* summed = embB   + (size_t)N_GENES * DD;       // BB*DD
  float* hbuf   = summed + (size_t)BB * DD;            // BB*DD
  float* ngen   = hbuf   + (size_t)BB * DD;            // BB
  int*   cnt    = (int*)(ngen + BB);                   // N_GENES
  int*   rowptr = cnt    + N_GENES;                    // N_GENES
  int*   cursor = rowptr + N_GENES;                    // N_GENES
  int*   ccol   = cursor + N_GENES;                    // NNZ_E
  float* cval   = (float*)(ccol + NNZ_E);              // NNZ_E

  const int edgeBlk  = (NNZ_E + 255) / 256;            // 2500
  const int rowBlk   = (N_GENES + 255) / 256;          // 79
  const int spmmBlk  = N_GENES / 8;                    // 2500 (one wave per row, 8 waves/block)
  const int gemmGenB = (N_GENES / 32) * 16 / 8;        // 1250 blocks
  const int gemmMlpB = (BB / 32) * 16 / 8;             // 1024 blocks
  const int gathBlk  = (BB * 64) / 256;                // 4096

  // --- Build CSR on device (deterministic work, rebuilt each call) ---
  zero_i32<<<rowBlk, 256, 0, stream>>>(cnt, N_GENES);
  count_edges<<<edgeBlk, 256, 0, stream>>>(arow, cnt);
  scan_rows<<<1, 256, 0, stream>>>(cnt, rowptr, cursor);
  scatter_edges<<<edgeBlk, 256, 0, stream>>>(arow, acol, avals, cursor, ccol, cval);

  // --- GNN layer 0 ---
  spmm_csr<<<spmmBlk, 256, 0, stream>>>(rowptr, cnt, ccol, cval, pert, aggA);
  gemm_wmma_f32<<<gemmGenB, 256, 0, stream>>>(aggA, gk, nullptr, embB,
                                              /*relu=*/1, nullptr, nullptr);
  // --- GNN layer 1 ---
  spmm_csr<<<spmmBlk, 256, 0, stream>>>(rowptr, cnt, ccol, cval, embB, aggA);
  gemm_wmma_f32<<<gemmGenB, 256, 0, stream>>>(aggA, gk + DD * DD, nullptr, embB,
                                              /*relu=*/0, nullptr, nullptr);
  // --- Per-condition gather-sum ---
  gather_sum<<<gathBlk, 256, 0, stream>>>(cond_idx, cgi, cgm, embB, summed, ngen);
  // --- MLP layer 1 ---
  gemm_wmma_f32<<<gemmMlpB, 256, 0, stream>>>(summed, w1, b1, hbuf,
                                              /*relu=*/1, nullptr, nullptr);
  // --- MLP layer 2 with blend epilogue -> final output ---
  gemm_wmma_f32<<<gemmMlpB, 256, 0, stream>>>(hbuf, w2, b2, out,
                                              /*relu=*/1, summed, ngen);
}